// FFMLayer_21912923144773
// MI455X (gfx1250) — compile-verified
//
#include <hip/hip_runtime.h>
#include <math.h>

#define BATCH   16384
#define ND      13
#define NS      26
#define OD      1000
#define GF      39
#define ROW     312          // GF * 8 floats per v-row
#define ROWB    1248         // ROW * 4 bytes
#define NTILES  20           // ceil(312/16), padded N = 320
#define LPITCH  324          // LDS row pitch in floats (pad vs. bank conflicts)
#define SPB     16           // samples per block
#define THREADS 128

typedef __attribute__((ext_vector_type(2))) float v2f;
typedef __attribute__((ext_vector_type(8))) float v8f;

__global__ __launch_bounds__(THREADS)
void ffm_fused_kernel(const float* __restrict__ dense_x,
                      const int*   __restrict__ sparse_x,
                      const float* __restrict__ W,
                      const float* __restrict__ bias,
                      const float* __restrict__ V,
                      float* __restrict__ out)
{
    __shared__ float    sF[SPB * LPITCH];   // dense field_f tile [16 samples][324]
    __shared__ float    sV[16 * LPITCH];    // zero-padded v[:13] as [K=16][324]
    __shared__ unsigned sOff[SPB * NS];     // gathered v-row byte offsets
    __shared__ float    sPart[SPB * 8 * 8]; // per-(sample,q) {s4, sq4}

    const int tid = threadIdx.x;
    const int s0  = blockIdx.x * SPB;

    // ---- stage v[:13] into LDS, zero-padded to [16][324] ----
    for (int e = tid; e < 16 * LPITCH; e += THREADS) {
        const int k = e / LPITCH, n = e % LPITCH;
        float val = 0.0f;
        if (k < ND && n < ROW) val = V[k * ROW + n];
        sV[e] = val;
    }
    // ---- precompute gather row byte-offsets ----
    for (int e = tid; e < SPB * NS; e += THREADS) {
        const int s = e / NS, i = e % NS;
        const int sx = sparse_x[(s0 + s) * NS + i];
        sOff[e] = (unsigned)(ND + i * OD + sx) * (unsigned)ROWB;
    }
    __syncthreads();

    // ---- Stage A: dense einsum [16,13] @ [13,312] via V_WMMA_F32_16X16X4_F32 ----
    {
        const int lane = tid & 31;
        // wave id is wave-uniform: pin it to an SGPR so the nt-loop is scalar
        // (keeps EXEC trivially all-1s around the WMMAs, no per-lane loop mask)
        const int wave = __builtin_amdgcn_readfirstlane(tid >> 5);   // 0..3
        const int lm   = lane & 15;     // M for A / N for B,C
        const int half = lane >> 4;     // 0/1

        // A fragments: K padded 13 -> 16, split into 4 WMMA K=4 steps.
        // Layout: VGPR p holds K = kk*4 + 2*half + p, M = lane%16.
        v2f a[4];
        #pragma unroll
        for (int kk = 0; kk < 4; ++kk) {
            #pragma unroll
            for (int p = 0; p < 2; ++p) {
                const int kg = kk * 4 + 2 * half + p;
                a[kk][p] = (kg < ND) ? dense_x[(s0 + lm) * ND + kg] : 0.0f;
            }
        }
        for (int nt = wave; nt < NTILES; nt += 4) {
            const int nb = nt * 16 + lm;
            v8f c = {};
            #pragma unroll
            for (int kk = 0; kk < 4; ++kk) {
                const int k0 = kk * 4 + 2 * half;
                v2f b;
                b[0] = sV[k0 * LPITCH + nb];
                b[1] = sV[(k0 + 1) * LPITCH + nb];
                c = __builtin_amdgcn_wmma_f32_16x16x4_f32(
                        false, a[kk], false, b, (short)0, c, false, false);
            }
            // C layout: VGPR p -> M = p + 8*half, N = lane%16
            #pragma unroll
            for (int p = 0; p < 8; ++p)
                sF[(p + 8 * half) * LPITCH + nb] = c[p];
        }
    }
    __syncthreads();

    // ---- Stage B: L2-resident gather-accumulate (the real bottleneck) ----
    {
        const int sl = tid >> 3;   // local sample 0..15
        const int q  = tid & 7;    // chunk owner within sample

        unsigned offr[NS];
        #pragma unroll
        for (int i = 0; i < NS; ++i) offr[i] = sOff[sl * NS + i];

        const char* vbase = (const char*)V;
        float4 s4  = make_float4(0.f, 0.f, 0.f, 0.f);
        float4 sq4 = make_float4(0.f, 0.f, 0.f, 0.f);

        // 78 float4 chunks per sample; e4 parity == q parity -> fixed k-phase
        for (int e4 = q; e4 < ROW / 4; e4 += 8) {
            const int byt = e4 * 16;
            float4 f = *(const float4*)&sF[sl * LPITCH + e4 * 4];
            #pragma unroll
            for (int i = 0; i < NS; ++i) {
                const float4 g = *(const float4*)(vbase + offr[i] + byt);
                f.x += g.x; f.y += g.y; f.z += g.z; f.w += g.w;
            }
            s4.x += f.x;       s4.y += f.y;       s4.z += f.z;       s4.w += f.w;
            sq4.x += f.x * f.x; sq4.y += f.y * f.y; sq4.z += f.z * f.z; sq4.w += f.w * f.w;
        }
        float4* pp = (float4*)&sPart[(sl * 8 + q) * 8];
        pp[0] = s4;
        pp[1] = sq4;
    }
    __syncthreads();

    // ---- finisher: combine partials, linear part, sigmoid ----
    if (tid < SPB) {
        const int s = tid;
        float sk[8]  = {0.f, 0.f, 0.f, 0.f, 0.f, 0.f, 0.f, 0.f};
        float sqk[8] = {0.f, 0.f, 0.f, 0.f, 0.f, 0.f, 0.f, 0.f};
        #pragma unroll
        for (int q = 0; q < 8; ++q) {
            const int ph = (q & 1) * 4;
            const float* p = &sPart[(s * 8 + q) * 8];
            #pragma unroll
            for (int c = 0; c < 4; ++c) {
                sk[ph + c]  += p[c];
                sqk[ph + c] += p[4 + c];
            }
        }
        float inter = 0.0f;
        #pragma unroll
        for (int k = 0; k < 8; ++k) inter += sk[k] * sk[k] - sqk[k];
        inter *= 0.5f;

        float lin = bias[0];
        const float* dx = &dense_x[(s0 + s) * ND];
        #pragma unroll
        for (int d = 0; d < ND; ++d) lin += dx[d] * W[d];
        const int* sx = &sparse_x[(s0 + s) * NS];
        #pragma unroll
        for (int i = 0; i < NS; ++i) lin += W[ND + i * OD + sx[i]];

        const float z = lin + inter;
        out[s0 + s] = 1.0f / (1.0f + __expf(-z));
    }
}

extern "C" void kernel_launch(void* const* d_in, const int* in_sizes, int n_in,
                              void* d_out, int out_size, void* d_ws, size_t ws_size,
                              hipStream_t stream) {
    (void)in_sizes; (void)n_in; (void)out_size; (void)d_ws; (void)ws_size;
    const float* dense_x  = (const float*)d_in[0];
    const int*   sparse_x = (const int*)d_in[1];
    const float* W        = (const float*)d_in[2];
    const float* bias     = (const float*)d_in[3];
    const float* V        = (const float*)d_in[4];
    float* out = (float*)d_out;

    dim3 grid(BATCH / SPB);
    dim3 block(THREADS);
    hipLaunchKernelGGL(ffm_fused_kernel, grid, block, 0, stream,
                       dense_x, sparse_x, W, bias, V, out);
}